// Net_61753039782760
// MI455X (gfx1250) — compile-verified
//
#include <hip/hip_runtime.h>
#include <hip/hip_bf16.h>

typedef float v2f __attribute__((ext_vector_type(2)));
typedef float v8f __attribute__((ext_vector_type(8)));

#define DIN   128
#define HID   128
#define DOUT  64
#define SLOPE 0.01f
#define LNEPS 1e-5f

// ---------------------------------------------------------------------------
// zero-fill
// ---------------------------------------------------------------------------
__global__ void zero_kernel(float* __restrict__ p, long long n) {
    long long i = (long long)blockIdx.x * blockDim.x + threadIdx.x;
    if (i < n) p[i] = 0.0f;
}

// ---------------------------------------------------------------------------
// SpMM scatter:  out[src[e]] += vals[e] * x[dst[e]]   (128-wide rows)
// one wave32 per edge, float4 per lane, 4 f32 atomics per lane.
// Gathers hit L2 (feature matrix is 25.6MB << 192MB L2).
// ---------------------------------------------------------------------------
__global__ __launch_bounds__(256)
void spmm_kernel(const int* __restrict__ src, const int* __restrict__ dst,
                 const float* __restrict__ vals, const float* __restrict__ x,
                 float* __restrict__ out, int E) {
    const int wave = (int)((blockIdx.x * blockDim.x + threadIdx.x) >> 5);
    const int lane = threadIdx.x & 31;
    if (wave >= E) return;
    const int s = src[wave];
    const int d = dst[wave];
    const float v = vals[wave];
    const float4 p = ((const float4*)(x + (size_t)d * 128))[lane];
    float* o = out + (size_t)s * 128 + lane * 4;
    atomicAdd(o + 0, v * p.x);
    atomicAdd(o + 1, v * p.y);
    atomicAdd(o + 2, v * p.z);
    atomicAdd(o + 3, v * p.w);
}

// ---------------------------------------------------------------------------
// LayerNorm over 128-wide rows: one wave32 per row, float4 per lane,
// shfl_xor tree reductions (wave32 native).
// ---------------------------------------------------------------------------
__global__ __launch_bounds__(256)
void ln_kernel(const float* __restrict__ in, const float* __restrict__ g,
               const float* __restrict__ b, float* __restrict__ out, int N) {
    const int row  = (int)((blockIdx.x * blockDim.x + threadIdx.x) >> 5);
    const int lane = threadIdx.x & 31;
    if (row >= N) return;
    const float4 p = ((const float4*)(in + (size_t)row * 128))[lane];

    float s = p.x + p.y + p.z + p.w;
    #pragma unroll
    for (int m = 16; m > 0; m >>= 1) s += __shfl_xor(s, m, 32);
    const float mu = s * (1.0f / 128.0f);

    const float dx = p.x - mu, dy = p.y - mu, dz = p.z - mu, dw = p.w - mu;
    float q = dx * dx + dy * dy + dz * dz + dw * dw;
    #pragma unroll
    for (int m = 16; m > 0; m >>= 1) q += __shfl_xor(q, m, 32);
    const float rs = rsqrtf(q * (1.0f / 128.0f) + LNEPS);

    const float4 gg = ((const float4*)g)[lane];
    const float4 bb = ((const float4*)b)[lane];
    float4 o;
    o.x = dx * rs * gg.x + bb.x;
    o.y = dy * rs * gg.y + bb.y;
    o.z = dz * rs * gg.z + bb.z;
    o.w = dw * rs * gg.w + bb.w;
    ((float4*)(out + (size_t)row * 128))[lane] = o;
}

// ---------------------------------------------------------------------------
// WMMA fp32 GEMM:  out[M,NOUT] = epilogue(A[M,128] @ B[128,NOUT])
// One wave32 per 16x16 output tile; K=128 as 32x V_WMMA_F32_16X16X4_F32.
//
// Fragment layouts per CDNA5 ISA 7.12.2 (32-bit A 16x4, B 4x16, C/D 16x16):
//   A: lanes 0-15 hold row M=lane, VGPR0=K0 VGPR1=K1; lanes 16-31: K2/K3
//   B: lanes 0-15 hold col N=lane, VGPR0=K0 VGPR1=K1; lanes 16-31: K2/K3
//   C/D: VGPR v -> row m0 + v + (lane>=16 ? 8 : 0), col n0 + (lane&15)
//
// MODE 0: out = leaky(acc + bias)                       (first linear)
// MODE 1: out = leaky(resin + acc + bias)               (residual layers)
// MODE 2: out = acc + bias                              (final projection)
// ---------------------------------------------------------------------------
template <int MODE, int NOUT>
__global__ __launch_bounds__(32)
void gemm128_kernel(const float* __restrict__ A, const float* __restrict__ B,
                    const float* __restrict__ bias,
                    const float* __restrict__ resin,
                    float* __restrict__ out, int M) {
    const int lane = threadIdx.x & 31;
    const int hi   = lane >> 4;       // 0: lanes 0-15, 1: lanes 16-31
    const int l16  = lane & 15;
    const int m0   = blockIdx.x * 16;
    const int n0   = blockIdx.y * 16;

    const float* arow = A + (size_t)(m0 + l16) * 128;
    const float* bcol = B + n0 + l16;

    v8f acc = {};
    #pragma unroll
    for (int k0 = 0; k0 < 128; k0 += 4) {
        const int ka = k0 + (hi << 1);
        v2f a, b;
        a.x = arow[ka];
        a.y = arow[ka + 1];
        b.x = bcol[(size_t)ka * NOUT];
        b.y = bcol[(size_t)(ka + 1) * NOUT];
        acc = __builtin_amdgcn_wmma_f32_16x16x4_f32(
            /*neg_a=*/false, a, /*neg_b=*/false, b,
            /*c_mod=*/(short)0, acc, /*reuse_a=*/false, /*reuse_b=*/false);
    }

    const int col = n0 + l16;
    const float bv = bias[col];
    #pragma unroll
    for (int v = 0; v < 8; ++v) {
        const int row = m0 + hi * 8 + v;
        float r = acc[v] + bv;
        if (MODE == 1) r += resin[(size_t)row * NOUT + col];
        if (MODE != 2) r = (r > 0.0f) ? r : SLOPE * r;
        out[(size_t)row * NOUT + col] = r;
    }
}

// ---------------------------------------------------------------------------
// launcher
// ---------------------------------------------------------------------------
extern "C" void kernel_launch(void* const* d_in, const int* in_sizes, int n_in,
                              void* d_out, int out_size, void* d_ws, size_t ws_size,
                              hipStream_t stream) {
    const float* x        = (const float*)d_in[0];
    const float* vals     = (const float*)d_in[1];
    const float* W1       = (const float*)d_in[2];
    const float* b1       = (const float*)d_in[3];
    const float* res_ln_g = (const float*)d_in[4];
    const float* res_ln_b = (const float*)d_in[5];
    const float* res_W    = (const float*)d_in[6];
    const float* res_b    = (const float*)d_in[7];
    const float* ln2_g    = (const float*)d_in[8];
    const float* ln2_b    = (const float*)d_in[9];
    const float* W2       = (const float*)d_in[10];
    const float* b2       = (const float*)d_in[11];
    const int*   src      = (const int*)d_in[12];
    const int*   dst      = (const int*)d_in[13];

    const int N = in_sizes[0] / DIN;   // 50000
    const int E = in_sizes[1];         // 800000

    float* bufA = (float*)d_ws;                    // [N,128] spmm accumulator
    float* bufB = bufA + (size_t)N * 128;          // [N,128] hidden state h
    float* bufC = bufB + (size_t)N * 128;          // [N,128] LN scratch
    float* outp = (float*)d_out;                   // [N,64]

    const long long nElem = (long long)N * 128;
    const int zgrid   = (int)((nElem + 255) / 256);
    const int egrid   = (E + 7) / 8;       // 8 waves (edges) per 256-thr block
    const int lngrid  = (N + 7) / 8;       // 8 rows per 256-thr block
    const dim3 g128((unsigned)(N / 16), HID / 16);   // (3125, 8)
    const dim3 g64 ((unsigned)(N / 16), DOUT / 16);  // (3125, 4)

    // ---- block 1: h = spmm(x); h = leaky(h @ W1 + b1) ----
    zero_kernel<<<zgrid, 256, 0, stream>>>(bufA, nElem);
    spmm_kernel<<<egrid, 256, 0, stream>>>(src, dst, vals, x, bufA, E);
    gemm128_kernel<0, HID><<<g128, 32, 0, stream>>>(bufA, W1, b1, nullptr, bufB, N);

    // ---- residual layers: h = leaky(h + LN(h) @ res_W[i] + res_b[i]) ----
    for (int i = 0; i < 2; ++i) {
        ln_kernel<<<lngrid, 256, 0, stream>>>(bufB, res_ln_g + i * HID,
                                              res_ln_b + i * HID, bufC, N);
        gemm128_kernel<1, HID><<<g128, 32, 0, stream>>>(
            bufC, res_W + (size_t)i * HID * HID, res_b + i * HID, bufB, bufB, N);
    }

    // ---- block 2: h = spmm(h); out = LN(h) @ W2 + b2 ----
    zero_kernel<<<zgrid, 256, 0, stream>>>(bufA, nElem);
    spmm_kernel<<<egrid, 256, 0, stream>>>(src, dst, vals, bufB, bufA, E);
    ln_kernel<<<lngrid, 256, 0, stream>>>(bufA, ln2_g, ln2_b, bufC, N);
    gemm128_kernel<2, DOUT><<<g64, 32, 0, stream>>>(bufC, W2, b2, nullptr, outp, N);
}